// DSelectKGate_69037304316407
// MI455X (gfx1250) — compile-verified
//
#include <hip/hip_runtime.h>

// Per-row top-8 selection (ascending), B=2048 rows, N=32768 cols, fp32.
// HBM-bound target: 256 MiB @ 23.3 TB/s => ~11.5 us floor.
// Streaming loop redesigned for VOPD dual-issue: per float4 we do a 5-cswap
// sort4 + one-shot bitonic "merge sorted-4 into sorted-8 keep top-8"
// (4 fmax + 12 independent cswaps) instead of 4 serial 7-deep bubble inserts.
// This cuts streaming VALU from ~16 cyc/elem to ~6 cyc/elem, below the HBM
// roofline. Loads are b128 non-temporal (input read once, > L2 capacity),
// plus global_prefetch_b8 lookahead. Cross-lane reduction: wave32 shfl_xor
// bitonic top-8 butterflies + a 256 B LDS hop across the block's 8 waves.

typedef __attribute__((ext_vector_type(4))) float v4f;

#define TOPK   8
#define BLOCK  256
#define NWAVE  (BLOCK / 32)

__device__ __forceinline__ void cswap(float& a, float& b) {
  // min -> a, max -> b (branchless; min/max pair dual-issues as VOPD)
  float lo = fminf(a, b);
  float hi = fmaxf(a, b);
  a = lo;
  b = hi;
}

// s ascending 8-list, t ascending 8-list -> s = ascending top-8 of union.
// m[i] = max(s[i], t[7-i]) is bitonic; 3-stage network sorts it.
__device__ __forceinline__ void merge_top8(float (&s)[TOPK], const float (&t)[TOPK]) {
  float m[TOPK];
#pragma unroll
  for (int i = 0; i < TOPK; ++i) m[i] = fmaxf(s[i], t[TOPK - 1 - i]);
  cswap(m[0], m[4]); cswap(m[1], m[5]); cswap(m[2], m[6]); cswap(m[3], m[7]);
  cswap(m[0], m[2]); cswap(m[1], m[3]); cswap(m[4], m[6]); cswap(m[5], m[7]);
  cswap(m[0], m[1]); cswap(m[2], m[3]); cswap(m[4], m[5]); cswap(m[6], m[7]);
#pragma unroll
  for (int i = 0; i < TOPK; ++i) s[i] = m[i];
}

// Fold one float4 into the ascending top-8 list s.
// Equivalent to merging the sorted 4-list, padded with -inf to 8, into s:
//   Bp = [-inf,-inf,-inf,-inf, b0,b1,b2,b3]  (ascending)
//   m[i] = max(s[i], Bp[7-i])  => m[0..3] = max(s[0..3], b3..b0), m[4..7]=s[4..7]
// m is bitonic; the 3-stage (4,2,1) network restores ascending order.
__device__ __forceinline__ void fold4(float (&s)[TOPK], v4f v) {
  float b0 = v[0], b1 = v[1], b2 = v[2], b3 = v[3];
  // prefilter: if even the max of the 4 can't beat the running 8th-largest,
  // nothing to do (fmax tree fuses to v_max3 + v_max).
  float pre = fmaxf(fmaxf(b0, b1), fmaxf(b2, b3));
  if (pre > s[0]) {
    // sort4 ascending: 3 dependency levels, pairs dual-issue
    cswap(b0, b1); cswap(b2, b3);
    cswap(b0, b2); cswap(b1, b3);
    cswap(b1, b2);
    float m[TOPK];
    m[0] = fmaxf(s[0], b3);
    m[1] = fmaxf(s[1], b2);
    m[2] = fmaxf(s[2], b1);
    m[3] = fmaxf(s[3], b0);
    m[4] = s[4]; m[5] = s[5]; m[6] = s[6]; m[7] = s[7];
    cswap(m[0], m[4]); cswap(m[1], m[5]); cswap(m[2], m[6]); cswap(m[3], m[7]);
    cswap(m[0], m[2]); cswap(m[1], m[3]); cswap(m[4], m[6]); cswap(m[5], m[7]);
    cswap(m[0], m[1]); cswap(m[2], m[3]); cswap(m[4], m[5]); cswap(m[6], m[7]);
#pragma unroll
    for (int i = 0; i < TOPK; ++i) s[i] = m[i];
  }
}

__global__ __launch_bounds__(BLOCK) void topk8_rows_kernel(
    const float* __restrict__ x, float* __restrict__ out, int N) {
  const int row  = blockIdx.x;
  const int tid  = threadIdx.x;
  const int lane = tid & 31;
  const int wv   = tid >> 5;

  const v4f* __restrict__ xr =
      reinterpret_cast<const v4f*>(x + (size_t)row * (size_t)N);
  const int nvec = N >> 2;  // float4 count per row (8192)

  float s[TOPK];
#pragma unroll
  for (int i = 0; i < TOPK; ++i) s[i] = -__builtin_inff();

  // --- streaming pass: unroll x2, both b128 NT loads in flight, prefetch ---
  int i = tid;
  for (; i + BLOCK < nvec; i += 2 * BLOCK) {
    v4f va = __builtin_nontemporal_load(&xr[i]);
    v4f vb = __builtin_nontemporal_load(&xr[i + BLOCK]);
    if (i + 8 * BLOCK < nvec) {
      __builtin_prefetch(&xr[i + 8 * BLOCK], 0, 0);  // global_prefetch_b8
    }
    fold4(s, va);
    fold4(s, vb);
  }
  if (i < nvec) {  // tail (not hit for N=32768, BLOCK=256)
    v4f va = __builtin_nontemporal_load(&xr[i]);
    fold4(s, va);
  }

  // --- wave32 butterfly: after 5 rounds every lane holds the wave's top-8 ---
#pragma unroll
  for (int off = 16; off >= 1; off >>= 1) {
    float t[TOPK];
#pragma unroll
    for (int j = 0; j < TOPK; ++j) t[j] = __shfl_xor(s[j], off, 32);
    merge_top8(s, t);
  }

  // --- cross-wave via tiny LDS hop (8 waves -> 8 lists of 8 floats) ---
  __shared__ float lds[NWAVE * TOPK];
  if (lane == 0) {
#pragma unroll
    for (int j = 0; j < TOPK; ++j) lds[wv * TOPK + j] = s[j];
  }
  __syncthreads();

  if (tid < 32) {
    float r[TOPK];
#pragma unroll
    for (int j = 0; j < TOPK; ++j)
      r[j] = (lane < NWAVE) ? lds[lane * TOPK + j] : -__builtin_inff();

    // merge the 8 lists held in lanes 0..7 (lanes 8..31 carry -inf padding)
#pragma unroll
    for (int off = NWAVE / 2; off >= 1; off >>= 1) {
      float t[TOPK];
#pragma unroll
      for (int j = 0; j < TOPK; ++j) t[j] = __shfl_xor(r[j], off, 32);
      merge_top8(r, t);
    }

    if (lane == 0) {
      v4f* o = reinterpret_cast<v4f*>(out + (size_t)row * TOPK);
      v4f lo, hi;
      lo[0] = r[0]; lo[1] = r[1]; lo[2] = r[2]; lo[3] = r[3];
      hi[0] = r[4]; hi[1] = r[5]; hi[2] = r[6]; hi[3] = r[7];
      o[0] = lo;  // ascending order, matching x.sort(axis=1)[:, -8:]
      o[1] = hi;
    }
  }
}

extern "C" void kernel_launch(void* const* d_in, const int* in_sizes, int n_in,
                              void* d_out, int out_size, void* d_ws, size_t ws_size,
                              hipStream_t stream) {
  const float* x = (const float*)d_in[0];
  // d_in[1] is k (== 8); kernel is specialized for k = 8 per the reference.
  float* out = (float*)d_out;

  const int B = out_size / TOPK;                 // 2048 rows
  const int N = in_sizes[0] / (B > 0 ? B : 1);   // 32768 cols

  topk8_rows_kernel<<<B, BLOCK, 0, stream>>>(x, out, N);
}